// Sampler_67542655697022
// MI455X (gfx1250) — compile-verified
//
#include <hip/hip_runtime.h>
#include <hip/hip_bf16.h>

// MI455X sampler: top-k/top-p multinomial sampling, B=128 rows x V=128256 vocab.
// Memory-bound (no matmul -> WMMA unreachable): HBM floor ~1 read + 1 write
// (pass 2 hits 192MB L2) ~ 131MB ~ 5.6us at 23.3 TB/s.
// Strategy: never sort 128k elements. top_k <= 64 so probs_sort has <=64
// nonzeros. Pass 1: per-thread maxima -> rank them (rank0 = global max M,
// rank63 = T1, a provable lower bound on the 64th-largest row element by the
// subset property). Pass 2: stable softmax denominator + gather {x >= T1}
// (~100 candidates for Gaussian rows) via LDS atomics. Then stable all-pairs
// rank -> sorted top-64, serial reference-order mask/renorm/inverse-CDF.

typedef __attribute__((ext_vector_type(4))) float v4f;

#define TPB  1024          // 32 wave32s, one workgroup per row
#define CAP  4096          // candidate capacity (Gaussian rows yield ~100)

__device__ __forceinline__ float wave_sum(float v) {
#pragma unroll
  for (int o = 16; o > 0; o >>= 1) v += __shfl_xor(v, o, 32);
  return v;
}

__global__ __launch_bounds__(TPB) void sampler_topk_topp_kernel(
    const float* __restrict__ logits,   // [B, V]
    const float* __restrict__ temps,    // [B]
    const int*   __restrict__ top_ks,   // [B]
    const float* __restrict__ top_ps,   // [B]
    const float* __restrict__ us,       // [B]
    float*       __restrict__ out,      // [B ids | B success | B*V probs_sort]
    int B, int V)
{
  const int b    = blockIdx.x;
  const int tid  = threadIdx.x;
  const int lane = tid & 31;
  const int wid  = tid >> 5;

  __shared__ float sh_tmax[TPB];
  __shared__ float sh_red[TPB / 32];
  __shared__ float sh_M, sh_T1, sh_S;
  __shared__ int   sh_cnt;
  __shared__ float sh_cval[CAP];
  __shared__ int   sh_cidx[CAP];
  __shared__ float sh_tv[64];
  __shared__ int   sh_ti[64];
  __shared__ float sh_pn[64];
  __shared__ float sh_cdf[64];

  const float t    = temps[b];
  const float invT = 1.0f / t;
  const float topp = top_ps[b];
  const int   kk   = min(top_ks[b], 64);
  const float u    = us[b];

  const float* row  = logits + (size_t)b * (size_t)V;
  const v4f*   row4 = (const v4f*)row;
  const int    n4   = V >> 2;

  // ---- Pass 1: per-thread running max over strided b128 stream ----
  float m = -INFINITY;
  for (int i = tid; i < n4; i += TPB) {
    if (i + TPB < n4) __builtin_prefetch(row4 + i + TPB, 0, 0);
    v4f x = row4[i];
    m = fmaxf(m, fmaxf(fmaxf(x.x, x.y), fmaxf(x.z, x.w)));
  }
  for (int i = (n4 << 2) + tid; i < V; i += TPB) m = fmaxf(m, row[i]);
  sh_tmax[tid] = m;
  if (tid == 0) sh_cnt = 0;
  if (tid < 64) { sh_tv[tid] = -INFINITY; sh_ti[tid] = 0; }
  __syncthreads();

  // Rank my max among all 1024 thread maxima (unique via tid tie-break),
  // scanning LDS with b128 reads. rank 0 -> global max M; rank 63 -> T1.
  {
    const v4f* tm4 = (const v4f*)sh_tmax;
    int r = 0;
    for (int q = 0; q < TPB / 4; ++q) {
      v4f vk = tm4[q];
      const int k = q << 2;
      r += (int)((vk.x > m) || (vk.x == m && (k + 0) < tid));
      r += (int)((vk.y > m) || (vk.y == m && (k + 1) < tid));
      r += (int)((vk.z > m) || (vk.z == m && (k + 2) < tid));
      r += (int)((vk.w > m) || (vk.w == m && (k + 3) < tid));
    }
    if (r == 0)  sh_M  = m;
    if (r == 63) sh_T1 = m;
  }
  __syncthreads();

  const float M  = sh_M;
  const float T1 = sh_T1;

  // ---- Pass 2 (L2-resident re-read): softmax denom + candidate gather ----
  float s = 0.0f;
  for (int i = tid; i < n4; i += TPB) {
    v4f x = row4[i];
    s += expf((x.x - M) * invT);
    s += expf((x.y - M) * invT);
    s += expf((x.z - M) * invT);
    s += expf((x.w - M) * invT);
    const int base = i << 2;
    if (x.x >= T1) { int p = atomicAdd(&sh_cnt, 1); if (p < CAP) { sh_cval[p] = x.x; sh_cidx[p] = base + 0; } }
    if (x.y >= T1) { int p = atomicAdd(&sh_cnt, 1); if (p < CAP) { sh_cval[p] = x.y; sh_cidx[p] = base + 1; } }
    if (x.z >= T1) { int p = atomicAdd(&sh_cnt, 1); if (p < CAP) { sh_cval[p] = x.z; sh_cidx[p] = base + 2; } }
    if (x.w >= T1) { int p = atomicAdd(&sh_cnt, 1); if (p < CAP) { sh_cval[p] = x.w; sh_cidx[p] = base + 3; } }
  }
  for (int i = (n4 << 2) + tid; i < V; i += TPB) {
    float x = row[i];
    s += expf((x - M) * invT);
    if (x >= T1) { int p = atomicAdd(&sh_cnt, 1); if (p < CAP) { sh_cval[p] = x; sh_cidx[p] = i; } }
  }
  // Block-wide sum reduction: wave32 shfl_xor + cross-wave LDS.
  s = wave_sum(s);
  if (lane == 0) sh_red[wid] = s;
  __syncthreads();
  if (wid == 0) {
    float z = sh_red[lane];
    z = wave_sum(z);
    if (lane == 0) sh_S = z;
  }
  __syncthreads();

  const float invS = 1.0f / sh_S;
  const int   C    = min(sh_cnt, CAP);

  // ---- Stable all-pairs rank of candidates -> sorted top-64 ----
  {
    const v4f* cv4 = (const v4f*)sh_cval;
    const int  C4  = C >> 2;
    for (int j = tid; j < C; j += TPB) {
      float vj = sh_cval[j]; int ij = sh_cidx[j];
      int r = 0;
      for (int q = 0; q < C4; ++q) {
        v4f vk = cv4[q];
        const int k = q << 2;
        r += (int)((vk.x > vj) || (vk.x == vj && sh_cidx[k + 0] < ij));
        r += (int)((vk.y > vj) || (vk.y == vj && sh_cidx[k + 1] < ij));
        r += (int)((vk.z > vj) || (vk.z == vj && sh_cidx[k + 2] < ij));
        r += (int)((vk.w > vj) || (vk.w == vj && sh_cidx[k + 3] < ij));
      }
      for (int k = C4 << 2; k < C; ++k) {
        float vk = sh_cval[k];
        r += (int)((vk > vj) || (vk == vj && sh_cidx[k] < ij));
      }
      if (r < 64) { sh_tv[r] = vj; sh_ti[r] = ij; }
    }
  }
  __syncthreads();

  // ---- Serial (64-wide) mask / renorm / inverse-CDF sample, ref order ----
  if (tid == 0) {
    float cum = 0.0f;
    for (int i = 0; i < 64; ++i) {
      float pi = expf((sh_tv[i] - M) * invT) * invS;  // -inf slots -> 0
      cum += pi;
      bool msk = ((cum - pi) > topp) || (i >= kk);
      sh_pn[i] = msk ? 0.0f : pi;
    }
    float inv0 = 1.0f / sh_pn[0];   // rank-0 is never masked
    float cdf  = 0.0f;
    for (int i = 0; i < 64; ++i) {
      float pn = sh_pn[i] * inv0;
      sh_pn[i] = pn;
      cdf += pn;
      sh_cdf[i] = cdf;
    }
    float r = u * cdf;              // cdf == cdf_last; row tail adds 0
    int rank = 0;
    for (int i = 0; i < 64; ++i) rank += (sh_cdf[i] < r) ? 1 : 0;
    rank = min(rank, 63);
    out[b]     = (float)sh_ti[rank];  // batch_next_token_ids
    out[B + b] = 1.0f;                // success
  }
  __syncthreads();

  // ---- Emit probs_sort row: 64-entry head + NT-streamed zero tail ----
  float* oprobs = out + 2 * B + (size_t)b * (size_t)V;
  if (tid < 64) oprobs[tid] = sh_pn[tid];
  v4f z4 = (v4f)0.0f;
  v4f* zp = (v4f*)(oprobs + 64);          // 16B aligned for this shape
  const int nz4 = (V - 64) >> 2;
  for (int i = tid; i < nz4; i += TPB) {
    __builtin_nontemporal_store(z4, zp + i);
  }
  for (int i = 64 + (nz4 << 2) + tid; i < V; i += TPB) oprobs[i] = 0.0f;
}

extern "C" void kernel_launch(void* const* d_in, const int* in_sizes, int n_in,
                              void* d_out, int out_size, void* d_ws, size_t ws_size,
                              hipStream_t stream) {
  const float* logits = (const float*)d_in[0];
  const float* temps  = (const float*)d_in[1];
  const int*   topks  = (const int*)d_in[2];
  const float* topps  = (const float*)d_in[3];
  const float* u      = (const float*)d_in[4];

  const int B = in_sizes[1];
  const int V = in_sizes[0] / B;

  sampler_topk_topp_kernel<<<B, TPB, 0, stream>>>(
      logits, temps, topks, topps, u, (float*)d_out, B, V);
}